// ExpectedCalibrationErrorLoss_29643864277072
// MI455X (gfx1250) — compile-verified
//
#include <hip/hip_runtime.h>
#include <hip/hip_bf16.h>
#include <stdint.h>

// ECE on MI455X (gfx1250): 400MB single-pass stream -> HBM-bound, ~17us floor
// at 23.3 TB/s. One wave32 DMAs a contiguous 32-row x 100-col fp32 tile
// (12.8KB) into LDS via CDNA5 GLOBAL_LOAD_ASYNC_TO_LDS_B128 (25 x 512B
// coalesced chunks, non-temporal, double-buffered, fenced with
// s_wait_asynccnt), then THREAD-per-row processing from LDS: ~15 issued
// instr/row, under the memory roofline. Per-block LDS histogram via
// ds_add_f32, global f32 atomics to workspace, tiny finalize kernel.

#define NUM_CLASSES 100
#define NBINS 10
#define ROWS_PER_TILE 32
#define TILE_FLOATS (ROWS_PER_TILE * NUM_CLASSES)   // 3200
#define CHUNKS 25                                   // 25 x 512B = 12800B per tile
#define WAVES_PER_BLOCK 2
#define NEG_INF (-3.402823466e38f)
#define LOG2E 1.4426950408889634f

__global__ void ece_zero_ws(float* ws) {
    if (threadIdx.x < 32) ws[threadIdx.x] = 0.0f;
}

// One async chunk: lane L copies 16B; INST_OFFSET applies to BOTH the LDS and
// global side, so 25 chunks share one base-address register pair. The logits
// stream is touched exactly once -> non-temporal hint keeps it out of L2.
__device__ __forceinline__ void issue_async_chunk(unsigned lds_byte_addr,
                                                  unsigned long long gaddr,
                                                  int imm_off) {
    asm volatile("global_load_async_to_lds_b128 %0, %1, off offset:%2 th:TH_LOAD_NT"
                 :: "v"(lds_byte_addr), "v"(gaddr), "i"(imm_off)
                 : "memory");
}

__global__ void __launch_bounds__(32 * WAVES_PER_BLOCK)
ece_main(const float* __restrict__ outputs,
         const int* __restrict__ targets,
         float* __restrict__ ws,
         int N) {
    // 2 waves x 2 buffers x 12.8KB = 51.2KB staging + histogram.
    __shared__ __align__(16) float stage[WAVES_PER_BLOCK][2][TILE_FLOATS];
    __shared__ float hist[32];   // [0,10) cnt, [10,20) sum_p, [20,30) sum_corr

    const int tid  = threadIdx.x;
    const int w    = tid >> 5;
    const int lane = tid & 31;

    if (tid < 32) hist[tid] = 0.0f;
    __syncthreads();

    const long long ntiles  = ((long long)N + ROWS_PER_TILE - 1) / ROWS_PER_TILE;
    const long long tstride = (long long)gridDim.x * WAVES_PER_BLOCK;
    long long tile = (long long)blockIdx.x * WAVES_PER_BLOCK + w;

    // Flat shared address truncated to 32 bits == DS byte address on CDNA5.
    const unsigned ldsBase0 = (unsigned)(uintptr_t)&stage[w][0][0] + (unsigned)(lane * 16);
    const unsigned ldsBase1 = (unsigned)(uintptr_t)&stage[w][1][0] + (unsigned)(lane * 16);

    // Issue one tile's 25 async chunks. Returns true if the tile was full
    // (exactly 25 chunks issued -> s_wait_asynccnt 25 is a valid fence).
    auto issue_tile = [&](long long t, int buf) -> bool {
        const unsigned lds = buf ? ldsBase1 : ldsBase0;
        const unsigned long long g =
            (unsigned long long)(uintptr_t)(outputs + t * TILE_FLOATS) +
            (unsigned long long)(lane * 16);
        const long long rowsLeft = (long long)N - t * ROWS_PER_TILE;
        if (rowsLeft >= ROWS_PER_TILE) {
            #pragma unroll
            for (int c = 0; c < CHUNKS; ++c) {
                issue_async_chunk(lds, g, c * 512);
            }
            return true;
        } else {
            const int validBytes = (int)rowsLeft * (NUM_CLASSES * 4);
            #pragma unroll
            for (int c = 0; c < CHUNKS; ++c) {
                if (c * 512 + lane * 16 < validBytes) {
                    issue_async_chunk(lds, g, c * 512);
                }
            }
            return false;
        }
    };

    if (tile < ntiles) {
        issue_tile(tile, 0);
    }

    int buf = 0;
    for (; tile < ntiles; tile += tstride) {
        const long long ntile = tile + tstride;
        bool nextFull = false;
        if (ntile < ntiles) {
            nextFull = issue_tile(ntile, buf ^ 1);
        }
        if (ntile < ntiles && nextFull) {
            // In-order completion: <=25 outstanding means current tile landed.
            asm volatile("s_wait_asynccnt 0x19" ::: "memory");
        } else {
            asm volatile("s_wait_asynccnt 0x0" ::: "memory");
        }

        // ---- thread-per-row processing from LDS ----
        const long long row = tile * ROWS_PER_TILE + lane;
        if (row < N) {
            const float* rowp = &stage[w][buf][lane * NUM_CLASSES];  // 400B, 16B-aligned

            float4 rv[CHUNKS];
            #pragma unroll
            for (int c = 0; c < CHUNKS; ++c) {
                rv[c] = ((const float4*)rowp)[c];
            }

            // row max (v_max3 chains; argmax-free: correctness checked by value)
            float m = NEG_INF;
            #pragma unroll
            for (int c = 0; c < CHUNKS; ++c) {
                m = fmaxf(m, fmaxf(fmaxf(rv[c].x, rv[c].y), fmaxf(rv[c].z, rv[c].w)));
            }

            // sum of exp(x - m) = sum of 2^(x*log2e - m*log2e)
            const float k = m * LOG2E;
            float s0 = 0.f, s1 = 0.f;
            #pragma unroll
            for (int c = 0; c < CHUNKS; ++c) {
                s0 += __builtin_amdgcn_exp2f(fmaf(rv[c].x, LOG2E, -k));
                s1 += __builtin_amdgcn_exp2f(fmaf(rv[c].y, LOG2E, -k));
                s0 += __builtin_amdgcn_exp2f(fmaf(rv[c].z, LOG2E, -k));
                s1 += __builtin_amdgcn_exp2f(fmaf(rv[c].w, LOG2E, -k));
            }
            const float s = s0 + s1;

            const int   t  = __builtin_nontemporal_load(&targets[row]);  // coalesced
            const float xt = rowp[t];               // one dynamic ds_load_b32
            const float et = __builtin_amdgcn_exp2f(fmaf(xt, LOG2E, -k));
            const float p  = __fdividef(et, s);
            const float correct = (xt == m) ? 1.0f : 0.0f;

            if (p > 0.0f) {  // p == 0 falls in no bin (matches reference)
                int bin = (int)ceilf(p * 10.0f) - 1;
                bin = bin < 0 ? 0 : (bin > NBINS - 1 ? NBINS - 1 : bin);
                atomicAdd(&hist[bin],             1.0f);
                atomicAdd(&hist[NBINS + bin],     p);
                atomicAdd(&hist[2 * NBINS + bin], correct);
            }
        }
        buf ^= 1;
    }

    __syncthreads();
    if (tid < 3 * NBINS) {
        atomicAdd(&ws[tid], hist[tid]);
    }
}

__global__ void ece_finalize(const float* __restrict__ ws, float* __restrict__ out) {
    const int l = threadIdx.x;
    float c = 0.f, sp = 0.f, sc = 0.f;
    if (l < NBINS) {
        c  = ws[l];
        sp = ws[NBINS + l];
        sc = ws[2 * NBINS + l];
    }
    float term = 0.f;
    if (c > 0.f) {
        term = c * fabsf(sp / c - sc / c);
    }
    float tot = c;
    #pragma unroll
    for (int off = 16; off > 0; off >>= 1) {
        term += __shfl_xor(term, off, 32);
        tot  += __shfl_xor(tot,  off, 32);
    }
    if (l == 0) {
        out[0] = (tot > 0.f) ? (term / tot) : 0.f;
    }
}

extern "C" void kernel_launch(void* const* d_in, const int* in_sizes, int n_in,
                              void* d_out, int out_size, void* d_ws, size_t ws_size,
                              hipStream_t stream) {
    const float* outputs = (const float*)d_in[0];
    const int*   targets = (const int*)d_in[1];
    float*       out     = (float*)d_out;
    float*       ws      = (float*)d_ws;

    const int N = in_sizes[1];  // batch size (targets count)

    ece_zero_ws<<<1, 32, 0, stream>>>(ws);

    const long long ntiles = ((long long)N + ROWS_PER_TILE - 1) / ROWS_PER_TILE;
    long long grid = (ntiles + WAVES_PER_BLOCK - 1) / WAVES_PER_BLOCK;
    if (grid > 4096) grid = 4096;
    if (grid < 1) grid = 1;
    ece_main<<<(int)grid, 32 * WAVES_PER_BLOCK, 0, stream>>>(outputs, targets, ws, N);

    ece_finalize<<<1, 32, 0, stream>>>(ws, out);
}